// CIF_7155415515214
// MI455X (gfx1250) — compile-verified
//
#include <hip/hip_runtime.h>
#include <stdint.h>

#define NT1 256   // threads for scan kernel
#define NT2 128   // threads for gather kernel
#define CHUNK 8   // x-rows staged per async batch

// LDS byte offset of a pointer into a __shared__ array (generic -> addrspace(3) -> u32)
typedef __attribute__((address_space(3))) float lds_float_t;
__device__ __forceinline__ uint32_t lds_byteoff(const void* p) {
  return (uint32_t)(uintptr_t)(const lds_float_t*)p;
}

// ---------------------------------------------------------------------------
// Kernel 1: per-batch masked sum, loss, scaled cumsum, fire detection.
// One block per batch. Produces: wmain[T], firepos[<=T], wleft[<=T], outlen,
// per-batch |sum - tgt|, and out_len written (as float) into d_out tail.
// ---------------------------------------------------------------------------
__global__ __launch_bounds__(NT1)
void cif_scan_kernel(const float* __restrict__ prob,
                     const int*   __restrict__ x_len,
                     const int*   __restrict__ tgt_len,
                     float* __restrict__ wmain,
                     int*   __restrict__ firepos,
                     float* __restrict__ wleftv,
                     int*   __restrict__ outlen,
                     float* __restrict__ lossv,
                     float* __restrict__ outlen_f,
                     int T)
{
  extern __shared__ float sm[];
  float* sp   = sm;                    // T masked probs
  float* redf = sm + T;                // NT1 float scan buffer
  int*   redi = (int*)(sm + T + NT1);  // NT1 int scan buffer

  const int   b   = blockIdx.x;
  const int   tid = threadIdx.x;
  const int   xl  = x_len[b];
  const float tgt = (float)tgt_len[b];
  const float* pr = prob + (size_t)b * T;

  for (int t = tid; t < T; t += NT1)
    sp[t] = (t < xl) ? pr[t] : 0.0f;
  __syncthreads();

  const int chunk = (T + NT1 - 1) / NT1;
  const int t0 = tid * chunk;
  const int t1 = min(t0 + chunk, T);

  // thread-local partial sum, then block inclusive scan (Hillis-Steele)
  float lsum = 0.0f;
  for (int t = t0; t < t1; ++t) lsum += sp[t];
  redf[tid] = lsum;
  __syncthreads();
  for (int off = 1; off < NT1; off <<= 1) {
    float u = (tid >= off) ? redf[tid - off] : 0.0f;
    __syncthreads();
    redf[tid] += u;
    __syncthreads();
  }
  const float total  = redf[NT1 - 1];
  const float offset = redf[tid] - lsum;        // exclusive prefix (unscaled)
  const float alpha  = tgt / total;
  if (tid == 0) lossv[b] = fabsf(total - tgt);

  // pass 2: count fires in my chunk (threshold == 1.0 -> floor crossings)
  float run = offset;
  int nf = 0;
  for (int t = t0; t < t1; ++t) {
    const float Cu = run + sp[t];
    const float C  = Cu  * alpha;
    const float Cp = run * alpha;
    if ((floorf(C) != floorf(Cp)) && (t < xl)) ++nf;
    run = Cu;
  }
  redi[tid] = nf;
  __syncthreads();
  for (int off = 1; off < NT1; off <<= 1) {
    int u = (tid >= off) ? redi[tid - off] : 0;
    __syncthreads();
    redi[tid] += u;
    __syncthreads();
  }
  int r = redi[tid] - nf;                       // exclusive fire-rank base
  const int totfires = redi[NT1 - 1];
  if (tid == 0) {
    outlen[b]   = totfires;
    outlen_f[b] = (float)totfires;
  }

  // pass 3: emit weights + fire positions
  run = offset;
  float* wm = wmain   + (size_t)b * T;
  int*   fp = firepos + (size_t)b * T;
  float* wl = wleftv  + (size_t)b * T;
  for (int t = t0; t < t1; ++t) {
    const float Cu = run + sp[t];
    const float C  = Cu  * alpha;
    const float Cp = run * alpha;
    const float dq = floorf(C);
    const bool fire = (dq != floorf(Cp)) && (t < xl);
    wm[t] = fire ? (dq - Cp) : (sp[t] * alpha);
    if (fire) { fp[r] = t; wl[r] = C - dq; ++r; }
    run = Cu;
  }
}

// ---------------------------------------------------------------------------
// Kernel 2: gather per output row (deterministic inverse of the scatter).
// Row j of batch b = sum_{t in (p_{j-1}, p_j]} wmain[t]*x[t]
//                  + wleft[j-1]*x[p_{j-1}]                       (j>0)
//                  + (p_0==0 ? wleft[0]*x[0] : 0)                (j==0)
// x rows are staged into LDS with CDNA5 async copies (ASYNCcnt path).
// ---------------------------------------------------------------------------
__global__ __launch_bounds__(NT2)
void cif_gather_kernel(const float* __restrict__ x,
                       const float* __restrict__ wmain,
                       const int*   __restrict__ firepos,
                       const float* __restrict__ wleftv,
                       const int*   __restrict__ outlen,
                       float* __restrict__ out,
                       int T, int D)
{
  extern __shared__ float xs[];           // CHUNK * TILE floats
  const int TILE = NT2 * 4;               // floats per channel tile
  const int b   = blockIdx.y;
  const int j   = blockIdx.x;
  const int tid = threadIdx.x;

  float* orow = out + ((size_t)b * T + j) * (size_t)D;
  const int nfil = outlen[b];

  if (j >= nfil) {                        // masked rows: write zeros
    const float4 z = make_float4(0.f, 0.f, 0.f, 0.f);
    for (int c = tid * 4; c < D; c += TILE)
      *(float4*)(orow + c) = z;
    return;
  }

  const int* fp  = firepos + (size_t)b * T;
  const int pj   = fp[j];
  const int pjm1 = (j == 0) ? -1 : fp[j - 1];
  const int t_lo = pjm1 + 1;
  const int t_hi = pj;
  const float* xb = x + (size_t)b * T * (size_t)D;
  const float* wm = wmain + (size_t)b * T;

  for (int cbase = 0; cbase < D; cbase += TILE) {
    const int  c   = cbase + tid * 4;
    const bool act = (c < D);
    float4 acc = make_float4(0.f, 0.f, 0.f, 0.f);

    for (int tc = t_lo; tc <= t_hi; tc += CHUNK) {
      const int nr = min(CHUNK, t_hi - tc + 1);
      if (act) {
        for (int rr = 0; rr < nr; ++rr) {
          const float* gp = xb + (size_t)(tc + rr) * D + c;
          const uint32_t lo = lds_byteoff(&xs[rr * TILE + tid * 4]);
          // CDNA5 async copy: 16B/lane global -> LDS, tracked by ASYNCcnt
          asm volatile("global_load_async_to_lds_b128 %0, %1, off"
                       :: "v"(lo), "v"(gp) : "memory");
        }
      }
      asm volatile("s_wait_asynccnt 0x0" ::: "memory");
      // each lane reads back exactly the bytes it requested -> no barrier
      if (act) {
        for (int rr = 0; rr < nr; ++rr) {
          const float  w  = wm[tc + rr];
          const float4 xv = *(const float4*)(&xs[rr * TILE + tid * 4]);
          acc.x += w * xv.x; acc.y += w * xv.y;
          acc.z += w * xv.z; acc.w += w * xv.w;
        }
      }
    }

    if (act) {
      if (j > 0) {                                   // w_left spill from previous fire
        const float  w  = wleftv[(size_t)b * T + (j - 1)];
        const float4 xv = *(const float4*)(xb + (size_t)pjm1 * D + c);
        acc.x += w * xv.x; acc.y += w * xv.y; acc.z += w * xv.z; acc.w += w * xv.w;
      }
      if (j == 0 && pj == 0) {                       // reference's fire-at-t0 edge case
        const float  w  = wleftv[(size_t)b * T];
        const float4 xv = *(const float4*)(xb + c);
        acc.x += w * xv.x; acc.y += w * xv.y; acc.z += w * xv.z; acc.w += w * xv.w;
      }
      *(float4*)(orow + c) = acc;
    }
  }
}

// ---------------------------------------------------------------------------
// Kernel 3: quantity loss mean
// ---------------------------------------------------------------------------
__global__ void cif_loss_kernel(const float* __restrict__ lossv, int B,
                                float* __restrict__ out_loss)
{
  if (threadIdx.x == 0 && blockIdx.x == 0) {
    float s = 0.f;
    for (int b = 0; b < B; ++b) s += lossv[b];
    *out_loss = s / (float)B;
  }
}

extern "C" void kernel_launch(void* const* d_in, const int* in_sizes, int n_in,
                              void* d_out, int out_size, void* d_ws, size_t ws_size,
                              hipStream_t stream)
{
  const float* x       = (const float*)d_in[0];
  const int*   x_len   = (const int*)  d_in[1];
  const float* prob    = (const float*)d_in[2];
  const int*   tgt_len = (const int*)  d_in[3];

  const int B = in_sizes[1];
  const int T = in_sizes[2] / B;
  const int D = in_sizes[0] / in_sizes[2];

  float* out       = (float*)d_out;                    // B*T*D
  float* outlen_fo = out + (size_t)B * T * D;          // B (out_len as float)
  float* loss_out  = outlen_fo + B;                    // 1

  // workspace partition (all 4-byte types)
  float* wmain   = (float*)d_ws;                       // B*T
  int*   firepos = (int*)  (wmain   + (size_t)B * T);  // B*T
  float* wleftv  = (float*)(firepos + (size_t)B * T);  // B*T
  int*   outlen  = (int*)  (wleftv  + (size_t)B * T);  // B
  float* lossv   = (float*)(outlen  + B);              // B

  const size_t shm1 = (size_t)(T + NT1) * sizeof(float) + NT1 * sizeof(int);
  hipLaunchKernelGGL(cif_scan_kernel, dim3(B), dim3(NT1), shm1, stream,
                     prob, x_len, tgt_len, wmain, firepos, wleftv,
                     outlen, lossv, outlen_fo, T);

  const size_t shm2 = (size_t)CHUNK * NT2 * 4 * sizeof(float);
  hipLaunchKernelGGL(cif_gather_kernel, dim3(T, B), dim3(NT2), shm2, stream,
                     x, wmain, firepos, wleftv, outlen, out, T, D);

  hipLaunchKernelGGL(cif_loss_kernel, dim3(1), dim3(32), 0, stream,
                     lossv, B, loss_out);
}